// MPNN_27788438405233
// MI455X (gfx1250) — compile-verified
//
#include <hip/hip_runtime.h>
#include <hip/hip_bf16.h>
#include <math.h>

#define Hdim 32
#define Fdim 128
#define Gdim 256

typedef float v2f __attribute__((ext_vector_type(2)));
typedef float v8f __attribute__((ext_vector_type(8)));

// ---------------- degree / normalization ----------------

__global__ void degree_count_kernel(const int* __restrict__ tgt, float* __restrict__ degE, int E) {
    int e = blockIdx.x * blockDim.x + threadIdx.x;
    if (e < E) atomicAdd(&degE[tgt[e]], 1.0f);
}

__global__ void dinv_kernel(const float* __restrict__ degE, float* __restrict__ dinv, int N) {
    int n = blockIdx.x * blockDim.x + threadIdx.x;
    if (n < N) dinv[n] = rsqrtf(degE[n] + 1.0f);
}

// ---------------- fp32 WMMA GEMM: C[M x Nc] = A[M x K] @ W[K x Nc] (+ bias) ----------------
// Compile-time K/Nc -> fully unrolled, immediate-offset loads, no per-iter address math.
// One wave computes one 16x16 output tile via V_WMMA_F32_16X16X4_F32.
// A frag (16x4): lanes 0-15 hold K=k,k+1 ; lanes 16-31 hold K=k+2,k+3 (M = lane&15)
// B frag (4x16): same K striping across lane halves (N = lane&15)
// C/D (16x16):  VGPR v holds M = (lane>=16 ? 8 : 0) + v, N = lane&15
// M-tail: clamp row instead of predicating loads (garbage acc rows are never stored),
// keeping EXEC all-1s and the inner loop free of exec save/restore.

template <int K, int Nc, bool HasBias>
__global__ void gemm_wmma_f32_t(const float* __restrict__ A, const float* __restrict__ W,
                                const float* __restrict__ bias, float* __restrict__ C,
                                int M, int Mtiles) {
    const int lane = threadIdx.x & 31;
    const int wid  = threadIdx.x >> 5;
    const int tm   = blockIdx.x * 4 + wid;   // 4 waves per block, one m-tile each
    const int tn   = blockIdx.y;
    if (tm >= Mtiles) return;                // wave-uniform exit

    const int half = lane >> 4;              // 0 or 1
    const int l16  = lane & 15;
    int row = tm * 16 + l16;                 // A-frag row (clamped for loads)
    if (row >= M) row = M - 1;
    const int col = tn * 16 + l16;           // B-frag / output column

    const float* Ap = A + (size_t)row * K + half * 2;        // 8B aligned
    const float* Wp = W + (size_t)(half * 2) * Nc + col;

    v8f acc = {};
#pragma unroll
    for (int k = 0; k < K; k += 4) {
        v2f a = *(const v2f*)(Ap + k);       // global_load_b64, immediate offset
        v2f b;
        b.x = Wp[(size_t)k * Nc];            // global_load_b32 x2, immediate offsets
        b.y = Wp[(size_t)k * Nc + Nc];
        acc = __builtin_amdgcn_wmma_f32_16x16x4_f32(
                  /*neg_a=*/false, a, /*neg_b=*/false, b,
                  /*c_mod=*/(short)0, acc, /*reuse_a=*/false, /*reuse_b=*/false);
    }

    const float bv = HasBias ? bias[col] : 0.0f;
#pragma unroll
    for (int v = 0; v < 8; ++v) {
        int r = tm * 16 + half * 8 + v;
        if (r < M) C[(size_t)r * Nc + col] = acc[v] + bv;
    }
}

// ---------------- edge scatter (GCN normalized messages) ----------------
// one wave32 per edge, one lane per feature: 128B coalesced gather + f32 atomic scatter (L2 resident)

__global__ void edge_agg_gcn_kernel(const int* __restrict__ src, const int* __restrict__ tgt,
                                    const float* __restrict__ dinv, const float* __restrict__ xw,
                                    float* __restrict__ agg, int E) {
    int e = blockIdx.x * (blockDim.x >> 5) + (threadIdx.x >> 5);
    int f = threadIdx.x & 31;
    if (e >= E) return;
    int s = src[e], t = tgt[e];
    float coef = dinv[s] * dinv[t];
    atomicAdd(&agg[(size_t)t * Hdim + f], coef * xw[(size_t)s * Hdim + f]);
}

// out = relu(agg + xw/deg + b)
__global__ void gcn_finish_kernel(const float* __restrict__ agg, const float* __restrict__ xw,
                                  const float* __restrict__ degE, const float* __restrict__ b,
                                  float* __restrict__ h, int N) {
    int i = blockIdx.x * blockDim.x + threadIdx.x;
    if (i >= N * Hdim) return;
    int n = i >> 5, f = i & 31;
    float deg = degE[n] + 1.0f;
    float v = agg[i] + xw[i] / deg + b[f];
    h[i] = fmaxf(v, 0.0f);
}

// ---------------- edge-attr gather: rs = hl[src, attr]; sum into tgt ----------------

__global__ void edge_agg_rs_kernel(const int* __restrict__ src, const int* __restrict__ tgt,
                                   const int* __restrict__ attr, const float* __restrict__ hl,
                                   float* __restrict__ agg, int E) {
    int e = blockIdx.x * (blockDim.x >> 5) + (threadIdx.x >> 5);
    int f = threadIdx.x & 31;
    if (e >= E) return;
    int s = src[e], t = tgt[e], a = attr[e];
    atomicAdd(&agg[(size_t)t * Hdim + f], hl[(size_t)s * (4 * Hdim) + a * Hdim + f]);
}

// ---------------- graph pooling ----------------

__global__ void gcount_kernel(const int* __restrict__ batch, float* __restrict__ gcnt, int N) {
    int n = blockIdx.x * blockDim.x + threadIdx.x;
    if (n < N) atomicAdd(&gcnt[batch[n]], 1.0f);
}

// g[batch[n]] += agg[n] / max(cnt[n],1)   (per-node mean over incoming edges, then graph-sum)
__global__ void pool_kernel(const int* __restrict__ batch, const float* __restrict__ agg,
                            const float* __restrict__ degE, float* __restrict__ g, int N) {
    int n = blockIdx.x * (blockDim.x >> 5) + (threadIdx.x >> 5);
    int f = threadIdx.x & 31;
    if (n >= N) return;
    float denom = fmaxf(degE[n], 1.0f);
    atomicAdd(&g[(size_t)batch[n] * Hdim + f], agg[(size_t)n * Hdim + f] / denom);
}

// ---------------- MLP head: sigmoid(relu(gmean @ Wm1 + bm1) @ Wm2 + bm2) ----------------
// one wave per graph row

__global__ void mlp_head_kernel(const float* __restrict__ g, const float* __restrict__ gcnt,
                                const float* __restrict__ Wm1, const float* __restrict__ bm1,
                                const float* __restrict__ Wm2, const float* __restrict__ bm2,
                                float* __restrict__ out) {
    __shared__ float gm[Hdim];
    int r = blockIdx.x;
    int lane = threadIdx.x;
    float den = fmaxf(gcnt[r], 1.0f);
    gm[lane] = g[(size_t)r * Hdim + lane] / den;
    __syncthreads();
    float t = bm1[lane];
#pragma unroll
    for (int k = 0; k < Hdim; ++k) t += gm[k] * Wm1[k * Hdim + lane];
    t = fmaxf(t, 0.0f);
    float p = t * Wm2[lane];
#pragma unroll
    for (int off = 16; off > 0; off >>= 1) p += __shfl_down(p, off, 32);
    if (lane == 0) out[r] = 1.0f / (1.0f + expf(-(p + bm2[0])));
}

// ---------------- host launch ----------------

extern "C" void kernel_launch(void* const* d_in, const int* in_sizes, int n_in,
                              void* d_out, int out_size, void* d_ws, size_t ws_size,
                              hipStream_t stream) {
    (void)n_in; (void)out_size; (void)ws_size;
    const float* x     = (const float*)d_in[0];
    const int*   eidx  = (const int*)d_in[1];
    const int*   eattr = (const int*)d_in[2];
    const int*   batch = (const int*)d_in[3];
    const float* W1    = (const float*)d_in[4];
    const float* b1    = (const float*)d_in[5];
    const float* W2    = (const float*)d_in[6];
    const float* b2    = (const float*)d_in[7];
    const float* Wl    = (const float*)d_in[8];
    const float* bl    = (const float*)d_in[9];
    const float* Wm1   = (const float*)d_in[10];
    const float* bm1   = (const float*)d_in[11];
    const float* Wm2   = (const float*)d_in[12];
    const float* bm2   = (const float*)d_in[13];
    float* out = (float*)d_out;

    const int N = in_sizes[0] / Fdim;
    const int E = in_sizes[1] / 2;
    const int* srcI = eidx;
    const int* tgtI = eidx + E;

    float* ws = (float*)d_ws;
    size_t off = 0;
    float* degE = ws + off; off += (size_t)N;
    float* dinv = ws + off; off += (size_t)N;
    float* B0   = ws + off; off += (size_t)N * Hdim;        // xw scratch
    float* B1   = ws + off; off += (size_t)N * Hdim;        // agg scratch
    float* B2   = ws + off; off += (size_t)N * Hdim;        // h
    float* B3   = ws + off; off += (size_t)N * 4 * Hdim;    // hl
    float* g    = ws + off; off += (size_t)Gdim * Hdim;
    float* gcnt = ws + off; off += (size_t)Gdim;

    const int EPB = 8;  // edges per 256-thread block (8 waves)
    const int Mtiles = (N + 15) / 16;
    const dim3 gridH((unsigned)((Mtiles + 3) / 4), Hdim / 16);        // Nc = 32
    const dim3 gridL((unsigned)((Mtiles + 3) / 4), (4 * Hdim) / 16);  // Nc = 128

    // degrees + dinv
    hipMemsetAsync(degE, 0, (size_t)N * sizeof(float), stream);
    degree_count_kernel<<<(E + 255) / 256, 256, 0, stream>>>(tgtI, degE, E);
    dinv_kernel<<<(N + 255) / 256, 256, 0, stream>>>(degE, dinv, N);

    // ---- conv1: h = relu(gcn(x, W1) + b1)
    gemm_wmma_f32_t<Fdim, Hdim, false><<<gridH, 128, 0, stream>>>(x, W1, nullptr, B0, N, Mtiles);
    hipMemsetAsync(B1, 0, (size_t)N * Hdim * sizeof(float), stream);
    edge_agg_gcn_kernel<<<(E + EPB - 1) / EPB, 256, 0, stream>>>(srcI, tgtI, dinv, B0, B1, E);
    gcn_finish_kernel<<<(N * Hdim + 255) / 256, 256, 0, stream>>>(B1, B0, degE, b1, B2, N);

    // ---- conv2: h = relu(gcn(h, W2) + b2)
    gemm_wmma_f32_t<Hdim, Hdim, false><<<gridH, 128, 0, stream>>>(B2, W2, nullptr, B0, N, Mtiles);
    hipMemsetAsync(B1, 0, (size_t)N * Hdim * sizeof(float), stream);
    edge_agg_gcn_kernel<<<(E + EPB - 1) / EPB, 256, 0, stream>>>(srcI, tgtI, dinv, B0, B1, E);
    gcn_finish_kernel<<<(N * Hdim + 255) / 256, 256, 0, stream>>>(B1, B0, degE, b2, B2, N);

    // ---- hl = h @ Wl + bl   (N x 128, viewed as N x 4 x 32)
    gemm_wmma_f32_t<Hdim, 4 * Hdim, true><<<gridL, 128, 0, stream>>>(B2, Wl, bl, B3, N, Mtiles);

    // ---- rs aggregation (edge-attr-selected rows, summed into tgt)
    hipMemsetAsync(B1, 0, (size_t)N * Hdim * sizeof(float), stream);
    edge_agg_rs_kernel<<<(E + EPB - 1) / EPB, 256, 0, stream>>>(srcI, tgtI, eattr, B3, B1, E);

    // ---- graph pooling
    hipMemsetAsync(g, 0, (size_t)Gdim * Hdim * sizeof(float), stream);
    hipMemsetAsync(gcnt, 0, (size_t)Gdim * sizeof(float), stream);
    gcount_kernel<<<(N + 255) / 256, 256, 0, stream>>>(batch, gcnt, N);
    pool_kernel<<<(N + EPB - 1) / EPB, 256, 0, stream>>>(batch, B1, degE, g, N);

    // ---- MLP head + sigmoid
    mlp_head_kernel<<<Gdim, Hdim, 0, stream>>>(g, gcnt, Wm1, bm1, Wm2, bm2, out);
}